// PrecisionAttentionBlock_7645041787176
// MI455X (gfx1250) — compile-verified
//
#include <hip/hip_runtime.h>

typedef __attribute__((ext_vector_type(2))) float v2f;
typedef __attribute__((ext_vector_type(4))) float v4f;
typedef __attribute__((ext_vector_type(8))) float v8f;

#define N_SEQ 384
#define D_EMB 256
#define NT    767          // 2n-1
#define PLANE   98304L     // n*d
#define QPLANE  37748736L  // n*n*d

// workspace layout (float offsets)
#define WS_KRE 0L
#define WS_KIM 196352L
#define WS_IV  392704L
#define WS_XQ  589056L
#define WS_XK  785664L
#define WS_XV  982272L
#define WS_XR  1178880L
#define WS_DEN 1375488L
#define WS_S   1522944L
#define WS_EE  1621248L
#define WS_PP  1817856L

// d_out layout (float offsets): est, out, Q_ij, X_ij_hat_all
#define OUT_EST 0L
#define OUT_OUT 196608L
#define OUT_Q   393216L
#define OUT_XH  38141952L

__device__ __forceinline__ v8f wmma_f32(v2f a, v2f b, v8f c) {
  // V_WMMA_F32_16X16X4_F32 : D = A(16x4) x B(4x16) + C  (fp32, wave32)
  return __builtin_amdgcn_wmma_f32_16x16x4_f32(false, a, false, b, (short)0, c,
                                               false, false);
}

// ---------------------------------------------------------------------------
// Kernel P: K_exp / inverse-variance tables over (t_ji, dim)
// ---------------------------------------------------------------------------
__global__ void precompute_kernel(const float* __restrict__ t_v,
                                  const float* __restrict__ lambda1,
                                  const float* __restrict__ lOm,
                                  const float* __restrict__ lOm0,
                                  const float* __restrict__ lGa,
                                  const float* __restrict__ lC,
                                  float* __restrict__ ws) {
  int tt = blockIdx.x;       // 0..766
  int dim = threadIdx.x;     // 0..255
  float l1re = lambda1[dim & 127];
  float l1im = lambda1[128 + (dim & 127)];
  float lam_re = -fabsf(l1re);
  float lam_im = (dim < 128) ? l1im : -l1im;
  float tji = (tt < N_SEQ - 1) ? -t_v[N_SEQ - 1 - tt] : t_v[tt - (N_SEQ - 1)];
  float at = fabsf(tji);
  float decay = expf(lam_re * at);
  float ang = lam_im * tji;
  long o = (long)tt * D_EMB + dim;
  ws[WS_KRE + o] = decay * cosf(ang);
  ws[WS_KIM + o] = decay * sinf(ang);
  float Om  = fabsf(lOm[dim]);
  float Om0 = fabsf(lOm0[dim]);
  float Ga  = fabsf(lGa[dim]);
  float Cc  = lC[dim];
  float eat = expf(2.0f * lam_re * at);
  float var = Cc * Cc * (Om / (-2.0f * lam_re + 1e-6f) * (1.0f - eat) + Om0 * eat)
              + Ga + 1e-6f;
  ws[WS_IV + o] = 1.0f / var;
}

// ---------------------------------------------------------------------------
// Kernel G: X_{q,k,v,r} = cmatmul(W_*, X) via WMMA f32 16x16x4.
// One wave per 16x16 output tile; 4 partial accumulators for complex product.
// ---------------------------------------------------------------------------
__global__ void proj_kernel(const float* __restrict__ X,
                            const float* __restrict__ Wq,
                            const float* __restrict__ Wk,
                            const float* __restrict__ Wv,
                            const float* __restrict__ Wr,
                            float* __restrict__ ws) {
  int wave = threadIdx.x >> 5;
  int lane = threadIdx.x & 31;
  int m  = lane & 15;
  int hi = lane >> 4;
  int T = blockIdx.x * 4 + wave;         // tile 0..383 within projection
  int proj = blockIdx.y;                 // 0..3
  const float* W = (proj == 0) ? Wq : (proj == 1) ? Wk : (proj == 2) ? Wv : Wr;
  float* O = ws + ((proj == 0) ? WS_XQ : (proj == 1) ? WS_XK
                                       : (proj == 2) ? WS_XV : WS_XR);
  int row0 = (T & 15) * 16;              // output dim tile
  int col0 = (T >> 4) * 16;              // token tile
  const float* Wre = W;
  const float* Wim = W + (long)D_EMB * D_EMB;
  const float* Xre = X;
  const float* Xim = X + PLANE;

  v8f aA = {0,0,0,0,0,0,0,0}, aB = aA, aC = aA, aD = aA;
  long aoff = (long)(row0 + m) * D_EMB + 2 * hi;   // A: W[row][k..k+1]
  long boff = (long)(col0 + m) * D_EMB + 2 * hi;   // B: X[token][k..k+1]
  for (int kb = 0; kb < D_EMB; kb += 4) {
    v2f wre = *(const v2f*)(Wre + aoff + kb);
    v2f wim = *(const v2f*)(Wim + aoff + kb);
    v2f xre = *(const v2f*)(Xre + boff + kb);
    v2f xim = *(const v2f*)(Xim + boff + kb);
    aA = wmma_f32(wre, xre, aA);   // Re*Re
    aB = wmma_f32(wim, xim, aB);   // Im*Im
    aC = wmma_f32(wre, xim, aC);   // Re*Im
    aD = wmma_f32(wim, xre, aD);   // Im*Re
  }
  // D layout: VGPR v holds M = v + 8*hi, N = lane&15 -> 8 contiguous dims
  float* ore = O + (long)(col0 + m) * D_EMB + row0 + 8 * hi;
  float* oim = ore + PLANE;
  v4f r0 = {aA[0]-aB[0], aA[1]-aB[1], aA[2]-aB[2], aA[3]-aB[3]};
  v4f r1 = {aA[4]-aB[4], aA[5]-aB[5], aA[6]-aB[6], aA[7]-aB[7]};
  v4f i0 = {aC[0]+aD[0], aC[1]+aD[1], aC[2]+aD[2], aC[3]+aD[3]};
  v4f i1 = {aC[4]+aD[4], aC[5]+aD[5], aC[6]+aD[6], aC[7]+aD[7]};
  *(v4f*)(ore)     = r0;  *(v4f*)(ore + 4) = r1;
  *(v4f*)(oim)     = i0;  *(v4f*)(oim + 4) = i1;
}

// ---------------------------------------------------------------------------
// Kernel C1: block = one i.  Thread layout: jj = tid>>6 (4 j-slices),
// dg = tid&63 (4 contiguous dims each -> b128 loads/stores).
// Per (i,j): denom (dim reduction), stream X_ij_hat, accumulate S partials.
// Each wave lies entirely inside one j-slice, so wave-reduce + 2 LDS partials
// give the per-j dim-sum with one barrier pair per FOUR j values.
// ---------------------------------------------------------------------------
__global__ void pass1_kernel(float* __restrict__ ws, float* __restrict__ out) {
  int i = blockIdx.x;
  int tid = threadIdx.x;
  int jj = tid >> 6;          // 0..3 : j-slice
  int dg = tid & 63;          // dim group
  int dim0 = dg * 4;
  __shared__ float red[8];    // per-wave partials
  __shared__ float Sp[4][D_EMB];
  const float* Kre = ws + WS_KRE;
  const float* Kim = ws + WS_KIM;
  const float* IV  = ws + WS_IV;
  const float* Xq  = ws + WS_XQ;
  const float* Xk  = ws + WS_XK;
  const float* Xv  = ws + WS_XV;
  float* xh_re = out + OUT_XH;
  float* xh_im = out + OUT_XH + QPLANE;

  v4f xqre = *(const v4f*)(Xq + (long)i * D_EMB + dim0);
  v4f xqim = *(const v4f*)(Xq + PLANE + (long)i * D_EMB + dim0);
  v4f Sacc = {0,0,0,0};
  for (int jb = 0; jb < N_SEQ; jb += 4) {
    int j = jb + jj;
    long idx = (long)(i - j + N_SEQ - 1) * D_EMB + dim0;
    v4f kre = *(const v4f*)(Kre + idx);
    v4f kim = *(const v4f*)(Kim + idx);
    v4f xvre = *(const v4f*)(Xv + (long)j * D_EMB + dim0);
    v4f xvim = *(const v4f*)(Xv + PLANE + (long)j * D_EMB + dim0);
    v4f xhre = kre * xvre - kim * xvim;
    v4f xhim = kre * xvim + kim * xvre;
    long o = ((long)i * N_SEQ + j) * D_EMB + dim0;
    *(v4f*)(xh_re + o) = xhre;
    *(v4f*)(xh_im + o) = xhim;
    v4f xkre = *(const v4f*)(Xk + (long)j * D_EMB + dim0);
    v4f xkim = *(const v4f*)(Xk + PLANE + (long)j * D_EMB + dim0);
    v4f rre = kre * xkre - kim * xkim - xqre;
    v4f rim = kre * xkim + kim * xkre - xqim;
    v4f p = *(const v4f*)(IV + idx);
    v4f pr = p * (rre * rre + rim * rim);
    float val = pr.x + pr.y + pr.z + pr.w;
    if (jb + 4 < N_SEQ)
      __builtin_prefetch(Xv + (long)(j + 4) * D_EMB + dim0, 0, 3);
    // wave32 reduce (each wave is inside one j-slice)
    for (int off = 16; off > 0; off >>= 1)
      val += __shfl_down(val, off, 32);
    if ((tid & 31) == 0) red[tid >> 5] = val;
    __syncthreads();
    float denom = 1.0f + red[2 * jj] + red[2 * jj + 1];   // NU = 1.0
    if (dg == 0) ws[WS_DEN + (long)i * N_SEQ + j] = denom;
    Sacc = Sacc + p * (1.0f / denom);
    __syncthreads();
  }
  // combine S partials across the 4 j-slices
  *(v4f*)(&Sp[jj][dim0]) = Sacc;
  __syncthreads();
  int dim = tid;  // 256 threads -> one dim each
  ws[WS_S + (long)i * D_EMB + dim] =
      Sp[0][dim] + Sp[1][dim] + Sp[2][dim] + Sp[3][dim];
}

// ---------------------------------------------------------------------------
// Kernel C2: same tiling; Q_ij b128 stream + est_v reduction + epilogue.
// ---------------------------------------------------------------------------
__global__ void pass2_kernel(float* __restrict__ ws, float* __restrict__ out) {
  int i = blockIdx.x;
  int tid = threadIdx.x;
  int jj = tid >> 6;
  int dg = tid & 63;
  int dim0 = dg * 4;
  __shared__ float EVre[4][D_EMB];
  __shared__ float EVim[4][D_EMB];
  const float* Kre = ws + WS_KRE;
  const float* Kim = ws + WS_KIM;
  const float* IV  = ws + WS_IV;
  const float* Xv  = ws + WS_XV;
  const float* Xr  = ws + WS_XR;
  const float* den = ws + WS_DEN;
  v4f Sv = *(const v4f*)(ws + WS_S + (long)i * D_EMB + dim0);
  v4f Sinv = {1.0f / Sv.x, 1.0f / Sv.y, 1.0f / Sv.z, 1.0f / Sv.w};
  float* qout = out + OUT_Q;
  v4f evre = {0,0,0,0}, evim = {0,0,0,0};
  for (int jb = 0; jb < N_SEQ; jb += 4) {
    int j = jb + jj;
    long idx = (long)(i - j + N_SEQ - 1) * D_EMB + dim0;
    v4f p = *(const v4f*)(IV + idx);
    float dnm = den[(long)i * N_SEQ + j];
    v4f q = p * (1.0f / dnm) * Sinv;
    long o = ((long)i * N_SEQ + j) * D_EMB + dim0;
    *(v4f*)(qout + o) = q;
    v4f kre = *(const v4f*)(Kre + idx);
    v4f kim = *(const v4f*)(Kim + idx);
    v4f xvre = *(const v4f*)(Xv + (long)j * D_EMB + dim0);
    v4f xvim = *(const v4f*)(Xv + PLANE + (long)j * D_EMB + dim0);
    evre = evre + q * (kre * xvre - kim * xvim);
    evim = evim + q * (kre * xvim + kim * xvre);
    if (jb + 4 < N_SEQ)
      __builtin_prefetch(Xv + (long)(j + 4) * D_EMB + dim0, 0, 3);
  }
  *(v4f*)(&EVre[jj][dim0]) = evre;
  *(v4f*)(&EVim[jj][dim0]) = evim;
  __syncthreads();
  int dim = tid;  // one dim per thread for the epilogue
  float sre = EVre[0][dim] + EVre[1][dim] + EVre[2][dim] + EVre[3][dim];
  float sim = EVim[0][dim] + EVim[1][dim] + EVim[2][dim] + EVim[3][dim];
  float xrre = Xr[(long)i * D_EMB + dim];
  float xrim = Xr[PLANE + (long)i * D_EMB + dim];
  float eere = 0.5f * sre + 0.5f * xrre;            // ALPHA = 0.5
  float eeim = 0.5f * sim + 0.5f * xrim;
  long o = (long)i * D_EMB + dim;
  ws[WS_EE + o] = eere;
  ws[WS_EE + PLANE + o] = eeim;
  // mat_exp = K_exp[:, 2n-1-(n+1) = 382]
  float mre = Kre[382L * D_EMB + dim];
  float mim = Kim[382L * D_EMB + dim];
  ws[WS_PP + o] = mre * eere - mim * eeim;
  ws[WS_PP + PLANE + o] = mre * eeim + mim * eere;
}

// ---------------------------------------------------------------------------
// Kernel D: est = W_e @ est_e, pred = W_p @ pred_p (both complex, WMMA f32);
//           out = 0.5*pred + 0.5*est.  One wave per 16x16 tile, both GEMMs.
// ---------------------------------------------------------------------------
__global__ void final_kernel(const float* __restrict__ We,
                             const float* __restrict__ Wp,
                             float* __restrict__ ws,
                             float* __restrict__ out) {
  int wave = threadIdx.x >> 5;
  int lane = threadIdx.x & 31;
  int m  = lane & 15;
  int hi = lane >> 4;
  int T = blockIdx.x * 4 + wave;    // 0..383
  int row0 = (T & 15) * 16;
  int col0 = (T >> 4) * 16;
  const float* Were = We;
  const float* Weim = We + (long)D_EMB * D_EMB;
  const float* Wpre = Wp;
  const float* Wpim = Wp + (long)D_EMB * D_EMB;
  const float* EEre = ws + WS_EE;
  const float* EEim = ws + WS_EE + PLANE;
  const float* PPre = ws + WS_PP;
  const float* PPim = ws + WS_PP + PLANE;

  v8f eA = {0,0,0,0,0,0,0,0}, eB = eA, eC = eA, eD = eA;
  v8f pA = eA, pB = eA, pC = eA, pD = eA;
  long aoff = (long)(row0 + m) * D_EMB + 2 * hi;
  long boff = (long)(col0 + m) * D_EMB + 2 * hi;
  for (int kb = 0; kb < D_EMB; kb += 4) {
    v2f were = *(const v2f*)(Were + aoff + kb);
    v2f weim = *(const v2f*)(Weim + aoff + kb);
    v2f wpre = *(const v2f*)(Wpre + aoff + kb);
    v2f wpim = *(const v2f*)(Wpim + aoff + kb);
    v2f bere = *(const v2f*)(EEre + boff + kb);
    v2f beim = *(const v2f*)(EEim + boff + kb);
    v2f bpre = *(const v2f*)(PPre + boff + kb);
    v2f bpim = *(const v2f*)(PPim + boff + kb);
    eA = wmma_f32(were, bere, eA);
    eB = wmma_f32(weim, beim, eB);
    eC = wmma_f32(were, beim, eC);
    eD = wmma_f32(weim, bere, eD);
    pA = wmma_f32(wpre, bpre, pA);
    pB = wmma_f32(wpim, bpim, pB);
    pC = wmma_f32(wpre, bpim, pC);
    pD = wmma_f32(wpim, bpre, pD);
  }
  float* estre = out + OUT_EST + (long)(col0 + m) * D_EMB + row0 + 8 * hi;
  float* estim = estre + PLANE;
  float* outre = out + OUT_OUT + (long)(col0 + m) * D_EMB + row0 + 8 * hi;
  float* outim = outre + PLANE;
  #pragma unroll
  for (int v = 0; v < 8; ++v) {
    float ere = eA[v] - eB[v], eim = eC[v] + eD[v];
    float pre = pA[v] - pB[v], pim = pC[v] + pD[v];
    estre[v] = ere;
    estim[v] = eim;
    outre[v] = 0.5f * pre + 0.5f * ere;   // DELTA = 0.5
    outim[v] = 0.5f * pim + 0.5f * eim;
  }
}

// ---------------------------------------------------------------------------
extern "C" void kernel_launch(void* const* d_in, const int* in_sizes, int n_in,
                              void* d_out, int out_size, void* d_ws, size_t ws_size,
                              hipStream_t stream) {
  const float* X    = (const float*)d_in[0];
  const float* t_v  = (const float*)d_in[1];
  const float* l1   = (const float*)d_in[2];
  const float* lOm  = (const float*)d_in[3];
  const float* lOm0 = (const float*)d_in[4];
  const float* lGa  = (const float*)d_in[5];
  const float* lC   = (const float*)d_in[6];
  const float* Wq   = (const float*)d_in[7];
  const float* Wk   = (const float*)d_in[8];
  const float* Wv   = (const float*)d_in[9];
  const float* Wr   = (const float*)d_in[10];
  const float* We   = (const float*)d_in[11];
  const float* Wp   = (const float*)d_in[12];
  float* ws  = (float*)d_ws;
  float* out = (float*)d_out;

  precompute_kernel<<<dim3(NT), dim3(D_EMB), 0, stream>>>(t_v, l1, lOm, lOm0,
                                                          lGa, lC, ws);
  proj_kernel<<<dim3(96, 4), dim3(128), 0, stream>>>(X, Wq, Wk, Wv, Wr, ws);
  pass1_kernel<<<dim3(N_SEQ), dim3(D_EMB), 0, stream>>>(ws, out);
  pass2_kernel<<<dim3(N_SEQ), dim3(D_EMB), 0, stream>>>(ws, out);
  final_kernel<<<dim3(96), dim3(128), 0, stream>>>(We, Wp, ws, out);
}